// AbstractTorchCircuit_51754355917582
// MI455X (gfx1250) — compile-verified
//
#include <hip/hip_runtime.h>

// ---------------------------------------------------------------------------
// Fused binary-tree of batched GEMMs for MI455X (gfx1250, wave32, WMMA).
//
//   leaves: h[f] = x[:, scope[f]] (outer) w_in[f]        (2048 x 64 per feat)
//   level l: p = h[a] .* h[b];  h_new[f] = p @ w_l[f]^T  (M=16-tile GEMMs)
//
// Kernel 1 (prep): bake all 255 node weight matrices to f16 in the exact
//   32x16 B-operand VGPR image (and leaf w_in pair-products in A-chunk
//   order) into d_ws.  ~2MB, fully L2-resident.
// Kernel 2 (tree): one workgroup (8 wave32) owns a 16-row batch tile and
//   runs ALL 8 levels in LDS: 64 live 16x64 f16 tiles (A-operand layout)
//   + per-wave temp tile.  v_wmma_f16_16x16x32_f16 (f16 C/D) so the result
//   tile needs zero conversion VALU before the LDS scatter.
// ---------------------------------------------------------------------------

typedef __attribute__((ext_vector_type(16))) _Float16 v16h;
typedef __attribute__((ext_vector_type(8)))  _Float16 v8h;

#define NWAVES 8
#define SLOT_H 1024   // halves per 16x64 tile (2KB)
#define NSLOTS 64     // live tiles after fused leaf+L0+L1 stage

// global feature index base per level (Fo = 128>>l): prefix sums
__device__ __constant__ const int kLvlOff[8] = {0, 128, 192, 224, 240, 248, 252, 254};
#define NFEAT_TOTAL 255
#define WB_HALVES   (NFEAT_TOTAL * 8 * 512)   // 255 feats * 8 operands * 512 halves
#define VP_HALVES   (128 * 64)                // 128 leaf pairs * (2q * 2hi * 16)

struct TreeArgs {
  const float* x;      // (2048, 1, 256) f32
  const int*   scope;  // (256, 1) i32  (identity in setup, honored anyway)
  const float* w_in;   // (256, 64, 1) f32
  const int*   idx[8]; // (F/2, 2) i32 per level
  const float* w[8];   // (F/2, 64, 64) f32 per level
  const _Float16* wB;  // baked B operands (ws)
  const _Float16* vp;  // baked leaf w_in pair products (ws)
  float*       out;    // (2048, 1, 64) f32
};

// Offset (in halves) of element (m,k) of a 16x64 tile stored as two
// consecutive 16x32 A-operand images (ISA 7.12.2, 16-bit A 16x32):
//   q = k>>5 selects the image; within image: lane = m + 16*((kk>>3)&1),
//   half j = ((kk>>4)&1)*8 + (kk&7); lane's 16 halves are contiguous (32B).
__device__ __forceinline__ int tile_off(int m, int k) {
  int q  = k >> 5;
  int kk = k & 31;
  int ln = m + 16 * ((kk >> 3) & 1);
  int j  = ((kk >> 4) & 1) * 8 + (kk & 7);
  return q * 512 + ln * 16 + j;
}

// ---------------- prep kernel: bake weights into operand images ------------
// Blocks 0..254: one node matrix each -> 8 B-operand images (f16).
// Block 255:     leaf w_in pair products in A-chunk order.
__global__ __launch_bounds__(256) void bake_weights_kernel(TreeArgs t, _Float16* wsH) {
  const int tid  = threadIdx.x;
  const int lane = tid & 31;
  const int wave = tid >> 5;
  const int g    = blockIdx.x;

  if (g < NFEAT_TOTAL) {
    // find (level, feature)
    int l = 0;
    while (l < 7 && g >= kLvlOff[l + 1]) ++l;
    const int f = g - kLvlOff[l];
    const float* wf = t.w[l] + f * 4096;      // (64 x 64) row-major [o][i]
    // wave -> operand (nt, q); B[i][o] = w[o][i] in 32x16 f16 operand image
    const int nt = wave >> 1, q = wave & 1;
    const int n = lane & 15, hi = lane >> 4;
    const float* wrow = wf + (nt * 16 + n) * 64 + q * 32;
    _Float16* dst = wsH + (size_t)g * 4096 + (nt * 2 + q) * 512 + lane * 16;
#pragma unroll
    for (int j = 0; j < 16; ++j) {
      int kk = ((j & 8) << 1) + 8 * hi + (j & 7);
      dst[j] = (_Float16)wrow[kk];
    }
  } else {
    // leaf pair products: vp[f0][q][hi][j] = w_in[a][k]*w_in[b][k]
    const int f0 = tid >> 1, q = tid & 1;     // 128 pairs * 2 chunks
    const int la = t.idx[0][2 * f0], lb = t.idx[0][2 * f0 + 1];
    const float* wa = t.w_in + la * 64 + q * 32;
    const float* wb = t.w_in + lb * 64 + q * 32;
    _Float16* dst = wsH + (size_t)WB_HALVES + f0 * 64 + q * 32;
#pragma unroll
    for (int hi = 0; hi < 2; ++hi)
#pragma unroll
      for (int j = 0; j < 16; ++j) {
        int kk = ((j & 8) << 1) + 8 * hi + (j & 7);
        dst[hi * 16 + j] = (_Float16)(wa[kk] * wb[kk]);
      }
  }
}

// ---------------- main kernel helpers --------------------------------------

// M=16, N=64, K=64 GEMM; B operands are pre-baked v16h images in global.
// f16 C/D: no conversion needed before storing the result tile.
__device__ __forceinline__ void gemm_tiles(v16h A0, v16h A1, const _Float16* wB8,
                                           int lane, v8h acc[4]) {
#pragma unroll
  for (int nt = 0; nt < 4; ++nt) {
    v16h b0 = *(const v16h*)(wB8 + (nt * 2 + 0) * 512 + lane * 16);
    acc[nt] = __builtin_amdgcn_wmma_f16_16x16x32_f16(
        false, A0, false, b0, (short)0, acc[nt], false, false);
    v16h b1 = *(const v16h*)(wB8 + (nt * 2 + 1) * 512 + lane * 16);
    acc[nt] = __builtin_amdgcn_wmma_f16_16x16x32_f16(
        false, A1, false, b1, (short)0, acc[nt], false, false);
  }
}

// Leaf pair product for level-0 feature f0: both leaf tiles are rank-1:
// p[m][k] = (x[m,a]*x[m,b]) * vp[f0][k], vp pre-baked in A-chunk order.
__device__ __forceinline__ void leaf_pair_A(const TreeArgs& t, int row0, int lane,
                                            int f0, v16h& A0, v16h& A1) {
  const int m = lane & 15, hi = lane >> 4;
  const int la = t.idx[0][2 * f0], lb = t.idx[0][2 * f0 + 1];
  const float s = t.x[(row0 + m) * 256 + t.scope[la]] *
                  t.x[(row0 + m) * 256 + t.scope[lb]];
  const _Float16 sh = (_Float16)s;
  const v16h vp0 = *(const v16h*)(t.vp + f0 * 64 + 0 * 32 + hi * 16);
  const v16h vp1 = *(const v16h*)(t.vp + f0 * 64 + 1 * 32 + hi * 16);
#pragma unroll
  for (int j = 0; j < 16; ++j) {
    A0[j] = vp0[j] * sh;
    A1[j] = vp1[j] * sh;
  }
}

// Load two LDS tiles (A-layout) and take the elementwise product.
__device__ __forceinline__ void slot_pair_A(const _Float16* pa, const _Float16* pb,
                                            int lane, v16h& A0, v16h& A1) {
  v16h a0 = *(const v16h*)(pa + lane * 16);
  v16h a1 = *(const v16h*)(pa + 512 + lane * 16);
  v16h b0 = *(const v16h*)(pb + lane * 16);
  v16h b1 = *(const v16h*)(pb + 512 + lane * 16);
  A0 = a0 * b0;
  A1 = a1 * b1;
}

// Scatter f16 D (16-bit C/D layout: lane n=L&15, hi=L>>4; half j <-> row
// m = 8*hi + j at column col) into the A-layout tile. Direct b16 stores.
__device__ __forceinline__ void store_tile(_Float16* dst, int lane, const v8h acc[4]) {
  int n = lane & 15, hi = lane >> 4;
#pragma unroll
  for (int nt = 0; nt < 4; ++nt) {
    int col = nt * 16 + n;
#pragma unroll
    for (int e = 0; e < 8; ++e) {
      int m = 8 * hi + e;
      dst[tile_off(m, col)] = acc[nt][e];
    }
  }
}

__device__ __forceinline__ void zero4(v8h acc[4]) {
  v8h z = {(_Float16)0, (_Float16)0, (_Float16)0, (_Float16)0,
           (_Float16)0, (_Float16)0, (_Float16)0, (_Float16)0};
  acc[0] = z; acc[1] = z; acc[2] = z; acc[3] = z;
}

// Compiler-only ordering fence for same-wave LDS store -> load (HW keeps
// DS ops from one wave in order; this just stops compiler reordering).
__device__ __forceinline__ void lds_fence() { asm volatile("" ::: "memory"); }

__launch_bounds__(256)
__global__ void tree_wmma_kernel(TreeArgs t) {
  extern __shared__ _Float16 smem[];
  _Float16* slots = smem;                                   // 64 * 2KB
  _Float16* temps = smem + NSLOTS * SLOT_H;                 // 8  * 2KB
  int* tab = (int*)(smem + (NSLOTS + NWAVES) * SLOT_H);     // 2 * 64 ints

  const int tid  = threadIdx.x;
  const int lane = tid & 31;
  const int wave = tid >> 5;
  const int row0 = blockIdx.x * 16;

  v16h A0, A1;
  v8h  acc[4];

  // ---- fused leaf + level 0 + level 1 (depth-first, 64 outputs) ----------
  for (int f1 = wave; f1 < 64; f1 += NWAVES) {
    int a1 = t.idx[1][2 * f1], b1 = t.idx[1][2 * f1 + 1];

    leaf_pair_A(t, row0, lane, a1, A0, A1);          // level-0 child a1
    zero4(acc);
    gemm_tiles(A0, A1, t.wB + (size_t)(kLvlOff[0] + a1) * 4096, lane, acc);
    store_tile(slots + f1 * SLOT_H, lane, acc);      // -> dest slot f1

    leaf_pair_A(t, row0, lane, b1, A0, A1);          // level-0 child b1
    zero4(acc);
    gemm_tiles(A0, A1, t.wB + (size_t)(kLvlOff[0] + b1) * 4096, lane, acc);
    store_tile(temps + wave * SLOT_H, lane, acc);    // -> wave temp

    lds_fence();
    slot_pair_A(slots + f1 * SLOT_H, temps + wave * SLOT_H, lane, A0, A1);
    zero4(acc);
    gemm_tiles(A0, A1, t.wB + (size_t)(kLvlOff[1] + f1) * 4096, lane, acc);
    store_tile(slots + f1 * SLOT_H, lane, acc);      // level-1 result in slot f1
  }
  if (tid < 64) tab[tid] = tid;                      // identity slot map
  __syncthreads();

  // ---- levels 2..7: in-place pair-combine with slot remap tables ----------
  int F = 32;
  for (int l = 2; l < 8; ++l) {
    int* cur = tab + ((l & 1) ? 64 : 0);
    int* nxt = tab + ((l & 1) ? 0 : 64);
    for (int f = wave; f < F; f += NWAVES) {
      int a  = t.idx[l][2 * f], b = t.idx[l][2 * f + 1];
      int sa = cur[a], sb = cur[b];
      slot_pair_A(slots + sa * SLOT_H, slots + sb * SLOT_H, lane, A0, A1);
      zero4(acc);
      gemm_tiles(A0, A1, t.wB + (size_t)(kLvlOff[l] + f) * 4096, lane, acc);
      store_tile(slots + sa * SLOT_H, lane, acc);    // pair writes only own slots
      if (lane == 0) nxt[f] = sa;
    }
    __syncthreads();
    F >>= 1;
  }

  // ---- emit: h is (1, B, 64); transpose(1,0,2) is flat-identical ----------
  const _Float16* res = slots + tab[0] * SLOT_H;     // l=7 wrote nxt = tab+0
  for (int e = tid; e < 16 * 64; e += 256) {
    int m = e >> 6, k = e & 63;
    t.out[(row0 + m) * 64 + k] = (float)res[tile_off(m, k)];
  }
}

extern "C" void kernel_launch(void* const* d_in, const int* in_sizes, int n_in,
                              void* d_out, int out_size, void* d_ws, size_t ws_size,
                              hipStream_t stream) {
  (void)in_sizes; (void)n_in; (void)out_size; (void)ws_size;
  _Float16* wsH = (_Float16*)d_ws;                  // ~2.02 MB used

  TreeArgs t;
  t.x     = (const float*)d_in[0];
  t.scope = (const int*)d_in[1];
  t.w_in  = (const float*)d_in[2];
  for (int l = 0; l < 8; ++l) {
    t.idx[l] = (const int*)d_in[3 + 2 * l];
    t.w[l]   = (const float*)d_in[4 + 2 * l];
  }
  t.wB  = wsH;
  t.vp  = wsH + WB_HALVES;
  t.out = (float*)d_out;

  // 1) bake weights (255 node matrices + leaf pair products) into ws
  bake_weights_kernel<<<NFEAT_TOTAL + 1, 256, 0, stream>>>(t, wsH);

  // 2) fused tree: 2048 batch rows / 16-row tiles = 128 workgroups of 8 wave32
  const size_t smem_bytes =
      (size_t)(NSLOTS + NWAVES) * SLOT_H * sizeof(_Float16) + 128 * sizeof(int);
  tree_wmma_kernel<<<128, 256, smem_bytes, stream>>>(t);
}